// DGLGAT_1468878815356
// MI455X (gfx1250) — compile-verified
//
#include <hip/hip_runtime.h>
#include <hip/hip_bf16.h>

// ---------------------------------------------------------------------------
// 2-layer GAT forward for MI455X (gfx1250, wave32, WMMA).
// GEMMs use V_WMMA_F32_16X16X4_F32 (full f32 precision, f32 accumulate).
// ---------------------------------------------------------------------------

typedef __attribute__((ext_vector_type(2))) float v2f;
typedef __attribute__((ext_vector_type(8))) float v8f;

// Problem constants (match reference).
static constexpr int N0 = 100000, N1 = 50000, N2 = 8000;
static constexpr int E0 = 600000, E1 = 80000;
static constexpr int F_IN = 256, HID = 64, NH = 4, NC = 47;
static constexpr int FEAT1 = NH * HID;      // 256
static constexpr int FEAT2 = NH * NC;       // 188
static constexpr int FEAT2_PAD = 192;       // padded to multiple of 64
static constexpr float NEG_SLOPE = 0.2f;

// ---------------------------------------------------------------------------
// Utility fills (deterministic init of workspace each call).
// ---------------------------------------------------------------------------
__global__ void fill_f32_kernel(float* __restrict__ p, float v, size_t n) {
  size_t i = (size_t)blockIdx.x * blockDim.x + threadIdx.x;
  size_t stride = (size_t)gridDim.x * blockDim.x;
  for (; i < n; i += stride) p[i] = v;
}

__global__ void fill_u32_kernel(unsigned* __restrict__ p, unsigned v, size_t n) {
  size_t i = (size_t)blockIdx.x * blockDim.x + threadIdx.x;
  size_t stride = (size_t)gridDim.x * blockDim.x;
  for (; i < n; i += stride) p[i] = v;
}

// ---------------------------------------------------------------------------
// WMMA f32 GEMM: C[M,N] = A[M,K] * B[K,N]
// Requirements: M % 16 == 0, N % 64 == 0, K % 4 == 0.
// One wave computes a 16x64 tile of C (4 accumulators, A fragment reused 4x).
// 8 waves / 256-thread block; flat wave-tile indexing with uniform tail guard.
// ---------------------------------------------------------------------------
__global__ __launch_bounds__(256) void wmma_gemm_f32_kernel(
    const float* __restrict__ A, const float* __restrict__ B,
    float* __restrict__ C, int M, int N, int K) {
  const int lane = threadIdx.x & 31;
  const int wave = threadIdx.x >> 5;
  const int tiles_n = N >> 6;                       // 64-wide column tiles
  const int total_tiles = (M >> 4) * tiles_n;
  const int wt = blockIdx.x * 8 + wave;
  if (wt >= total_tiles) return;                    // uniform per wave
  const int rt = wt / tiles_n;
  const int ct = wt - rt * tiles_n;
  const int row0 = rt << 4;
  const int col0 = ct << 6;

  const int mh = lane & 15;                         // M index (A) / N index (B)
  const int khalf = lane >> 4;                      // 0: K={k0,k0+1}, 1: K={k0+2,k0+3}

  v8f acc0 = {}, acc1 = {}, acc2 = {}, acc3 = {};

  const float* aptr = A + (size_t)(row0 + mh) * K + 2 * khalf;
  const float* bbase = B + (size_t)(2 * khalf) * N + col0 + mh;

  for (int k0 = 0; k0 < K; k0 += 4) {
    v2f a;
    a.x = aptr[k0];
    a.y = aptr[k0 + 1];
    const float* bp = bbase + (size_t)k0 * N;
    v2f b0, b1, b2, b3;
    b0.x = bp[0];       b0.y = bp[N];
    b1.x = bp[16];      b1.y = bp[N + 16];
    b2.x = bp[32];      b2.y = bp[N + 32];
    b3.x = bp[48];      b3.y = bp[N + 48];
    acc0 = __builtin_amdgcn_wmma_f32_16x16x4_f32(false, a, false, b0, (short)0, acc0, false, false);
    acc1 = __builtin_amdgcn_wmma_f32_16x16x4_f32(false, a, false, b1, (short)0, acc1, false, false);
    acc2 = __builtin_amdgcn_wmma_f32_16x16x4_f32(false, a, false, b2, (short)0, acc2, false, false);
    acc3 = __builtin_amdgcn_wmma_f32_16x16x4_f32(false, a, false, b3, (short)0, acc3, false, false);
  }

  // C/D layout: VGPR i -> lanes 0-15: M=i, N=lane; lanes 16-31: M=i+8, N=lane-16
  float* cp = C + (size_t)row0 * N + col0;
  const int m_base = khalf * 8;
  const int n = mh;
#pragma unroll
  for (int i = 0; i < 8; ++i) {
    float* crow = cp + (size_t)(m_base + i) * N + n;
    crow[0]  = acc0[i];
    crow[16] = acc1[i];
    crow[32] = acc2[i];
    crow[48] = acc3[i];
  }
}

// ---------------------------------------------------------------------------
// Pad W2 [256,188] -> Bp [256,192] (zero tail columns).
// ---------------------------------------------------------------------------
__global__ void pad_w2_kernel(const float* __restrict__ W2, float* __restrict__ Bp) {
  size_t i = (size_t)blockIdx.x * blockDim.x + threadIdx.x;
  size_t total = (size_t)F_IN * FEAT2_PAD;   // K for layer2 is 256 = H*HID
  size_t stride = (size_t)gridDim.x * blockDim.x;
  for (; i < total; i += stride) {
    int k = (int)(i / FEAT2_PAD);
    int j = (int)(i % FEAT2_PAD);
    Bp[i] = (j < FEAT2) ? W2[(size_t)k * FEAT2 + j] : 0.0f;
  }
}

// ---------------------------------------------------------------------------
// Attention coefficients: el[n,h] = feat[n,h,:]·al[h,:] for n < nsrc,
//                         er[n,h] = feat[n,h,:]·ar[h,:] for n < ndst.
// ---------------------------------------------------------------------------
__global__ void attn_coeff_kernel(const float* __restrict__ feat, int ld,
                                  const float* __restrict__ al,
                                  const float* __restrict__ ar,
                                  float* __restrict__ el, float* __restrict__ er,
                                  int nsrc, int ndst, int nh, int d) {
  size_t i = (size_t)blockIdx.x * blockDim.x + threadIdx.x;
  size_t total = (size_t)nsrc * nh;
  size_t stride = (size_t)gridDim.x * blockDim.x;
  for (; i < total; i += stride) {
    int n = (int)(i / nh);
    int h = (int)(i % nh);
    const float* f = feat + (size_t)n * ld + h * d;
    const float* wl = al + (size_t)h * d;
    const float* wr = ar + (size_t)h * d;
    float sl = 0.f, sr = 0.f;
#pragma unroll 4
    for (int c = 0; c < d; ++c) {
      float fv = f[c];
      sl = fmaf(fv, wl[c], sl);
      sr = fmaf(fv, wr[c], sr);
    }
    el[i] = sl;
    if (n < ndst) er[(size_t)n * nh + h] = sr;
  }
}

// ---------------------------------------------------------------------------
// Ordered-uint encoding for float atomicMax (monotonic over all finite floats).
// ---------------------------------------------------------------------------
__device__ __forceinline__ unsigned f32_key(float f) {
  unsigned u = __float_as_uint(f);
  return (u & 0x80000000u) ? ~u : (u | 0x80000000u);
}
__device__ __forceinline__ float key_f32(unsigned u) {
  return (u & 0x80000000u) ? __uint_as_float(u & 0x7fffffffu)
                           : __uint_as_float(~u);
}
static constexpr unsigned NEG_INF_KEY = 0x007fffffu;  // f32_key(-inf)

// Pass 1: e = leaky_relu(el[src]+er[dst]); stash e; segment max via atomicMax.
__global__ void edge_logit_max_kernel(const int* __restrict__ src,
                                      const int* __restrict__ dst,
                                      const float* __restrict__ el,
                                      const float* __restrict__ er,
                                      float* __restrict__ eval,
                                      unsigned* __restrict__ mkey,
                                      int E, int nh) {
  size_t i = (size_t)blockIdx.x * blockDim.x + threadIdx.x;
  size_t total = (size_t)E * nh;
  size_t stride = (size_t)gridDim.x * blockDim.x;
  for (; i < total; i += stride) {
    int e = (int)(i / nh);
    int h = (int)(i % nh);
    int sn = src[e], dn = dst[e];
    float v = el[(size_t)sn * nh + h] + er[(size_t)dn * nh + h];
    v = (v > 0.f) ? v : v * NEG_SLOPE;
    eval[i] = v;
    atomicMax(&mkey[(size_t)dn * nh + h], f32_key(v));
  }
}

// Pass 2: ee = exp(e - m[dst]); segment sum via atomicAdd. In-place on eval.
__global__ void edge_exp_sum_kernel(const int* __restrict__ dst,
                                    float* __restrict__ eval_ee,
                                    const unsigned* __restrict__ mkey,
                                    float* __restrict__ ssum, int E, int nh) {
  size_t i = (size_t)blockIdx.x * blockDim.x + threadIdx.x;
  size_t total = (size_t)E * nh;
  size_t stride = (size_t)gridDim.x * blockDim.x;
  for (; i < total; i += stride) {
    int e = (int)(i / nh);
    int h = (int)(i % nh);
    int dn = dst[e];
    float m = key_f32(mkey[(size_t)dn * nh + h]);
    float ee = expf(eval_ee[i] - m);
    eval_ee[i] = ee;
    atomicAdd(&ssum[(size_t)dn * nh + h], ee);
  }
}

// Pass 3: out[dst, f] += (ee/s[dst,h]) * feat[src, f]; one thread per edge×feat.
__global__ void edge_aggregate_kernel(const int* __restrict__ src,
                                      const int* __restrict__ dst,
                                      const float* __restrict__ ee,
                                      const float* __restrict__ ssum,
                                      const float* __restrict__ feat, int ldf,
                                      float* __restrict__ out, int ldo,
                                      long long E, int nh, int d) {
  const int F = nh * d;
  size_t i = (size_t)blockIdx.x * blockDim.x + threadIdx.x;
  size_t total = (size_t)E * F;
  size_t stride = (size_t)gridDim.x * blockDim.x;
  for (; i < total; i += stride) {
    size_t e = i / F;
    int f = (int)(i % F);
    int h = f / d;
    int sn = src[e], dn = dst[e];
    float alpha = ee[e * nh + h] / ssum[(size_t)dn * nh + h];
    atomicAdd(&out[(size_t)dn * ldo + f], alpha * feat[(size_t)sn * ldf + f]);
  }
}

// bias + ELU, in place: buf[n,f] = elu(buf[n,f] + bias[f])
__global__ void bias_elu_kernel(float* __restrict__ buf,
                                const float* __restrict__ bias,
                                int rows, int F) {
  size_t i = (size_t)blockIdx.x * blockDim.x + threadIdx.x;
  size_t total = (size_t)rows * F;
  size_t stride = (size_t)gridDim.x * blockDim.x;
  for (; i < total; i += stride) {
    int f = (int)(i % F);
    float v = buf[i] + bias[f];
    buf[i] = (v > 0.f) ? v : (expf(v) - 1.0f);
  }
}

// Final: out[n,c] = mean_h(agg2[n, h*NC + c] + b2[h*NC + c])
__global__ void head_mean_kernel(const float* __restrict__ agg2,
                                 const float* __restrict__ b2,
                                 float* __restrict__ out) {
  size_t i = (size_t)blockIdx.x * blockDim.x + threadIdx.x;
  size_t total = (size_t)N2 * NC;
  size_t stride = (size_t)gridDim.x * blockDim.x;
  for (; i < total; i += stride) {
    int n = (int)(i / NC);
    int c = (int)(i % NC);
    float s = 0.f;
#pragma unroll
    for (int h = 0; h < NH; ++h) {
      int f = h * NC + c;
      s += agg2[(size_t)n * FEAT2 + f] + b2[f];
    }
    out[i] = s * (1.0f / NH);
  }
}

// ---------------------------------------------------------------------------
// Host-side launcher.
// ---------------------------------------------------------------------------
static inline void* ws_take(char*& p, size_t bytes) {
  void* r = (void*)p;
  p += (bytes + 255) & ~(size_t)255;
  return r;
}

static inline int blocks_for(size_t total, int tpb = 256) {
  size_t b = (total + tpb - 1) / tpb;
  if (b > 1048576) b = 1048576;  // grid-stride handles the rest
  return (int)b;
}

extern "C" void kernel_launch(void* const* d_in, const int* in_sizes, int n_in,
                              void* d_out, int out_size, void* d_ws, size_t ws_size,
                              hipStream_t stream) {
  (void)in_sizes; (void)n_in; (void)out_size; (void)ws_size;

  const float* x   = (const float*)d_in[0];
  const float* W1  = (const float*)d_in[1];
  const float* al1 = (const float*)d_in[2];
  const float* ar1 = (const float*)d_in[3];
  const float* b1  = (const float*)d_in[4];
  const float* W2  = (const float*)d_in[5];
  const float* al2 = (const float*)d_in[6];
  const float* ar2 = (const float*)d_in[7];
  const float* b2  = (const float*)d_in[8];
  const int* src0  = (const int*)d_in[9];
  const int* dst0  = (const int*)d_in[10];
  const int* src1  = (const int*)d_in[11];
  const int* dst1  = (const int*)d_in[12];
  float* out = (float*)d_out;

  // Workspace layout.
  char* wp = (char*)d_ws;
  float*    feat1 = (float*)   ws_take(wp, (size_t)N0 * FEAT1 * 4);      // 102.4 MB
  float*    el1   = (float*)   ws_take(wp, (size_t)N0 * NH * 4);
  float*    er1   = (float*)   ws_take(wp, (size_t)N1 * NH * 4);
  unsigned* m1    = (unsigned*)ws_take(wp, (size_t)N1 * NH * 4);
  float*    s1    = (float*)   ws_take(wp, (size_t)N1 * NH * 4);
  float*    ee1   = (float*)   ws_take(wp, (size_t)E0 * NH * 4);
  float*    agg1  = (float*)   ws_take(wp, (size_t)N1 * FEAT1 * 4);      // 51.2 MB
  float*    Bp    = (float*)   ws_take(wp, (size_t)F_IN * FEAT2_PAD * 4);
  float*    feat2 = (float*)   ws_take(wp, (size_t)N1 * FEAT2_PAD * 4);  // 38.4 MB
  float*    el2   = (float*)   ws_take(wp, (size_t)N1 * NH * 4);
  float*    er2   = (float*)   ws_take(wp, (size_t)N2 * NH * 4);
  unsigned* m2    = (unsigned*)ws_take(wp, (size_t)N2 * NH * 4);
  float*    s2    = (float*)   ws_take(wp, (size_t)N2 * NH * 4);
  float*    ee2   = (float*)   ws_take(wp, (size_t)E1 * NH * 4);
  float*    agg2  = (float*)   ws_take(wp, (size_t)N2 * FEAT2 * 4);

  const int TPB = 256;

  // ---------------- Layer 1 ----------------
  // GEMM1: feat1 = x @ W1   (M=100000, N=256, K=256)
  {
    int tiles = (N0 / 16) * (FEAT1 / 64);
    wmma_gemm_f32_kernel<<<(tiles + 7) / 8, TPB, 0, stream>>>(x, W1, feat1, N0, FEAT1, F_IN);
  }
  attn_coeff_kernel<<<blocks_for((size_t)N0 * NH), TPB, 0, stream>>>(
      feat1, FEAT1, al1, ar1, el1, er1, N0, N1, NH, HID);

  fill_u32_kernel<<<blocks_for((size_t)N1 * NH), TPB, 0, stream>>>(m1, NEG_INF_KEY, (size_t)N1 * NH);
  fill_f32_kernel<<<blocks_for((size_t)N1 * NH), TPB, 0, stream>>>(s1, 0.f, (size_t)N1 * NH);
  fill_f32_kernel<<<blocks_for((size_t)N1 * FEAT1), TPB, 0, stream>>>(agg1, 0.f, (size_t)N1 * FEAT1);

  edge_logit_max_kernel<<<blocks_for((size_t)E0 * NH), TPB, 0, stream>>>(
      src0, dst0, el1, er1, ee1, m1, E0, NH);
  edge_exp_sum_kernel<<<blocks_for((size_t)E0 * NH), TPB, 0, stream>>>(
      dst0, ee1, m1, s1, E0, NH);
  edge_aggregate_kernel<<<blocks_for((size_t)E0 * FEAT1), TPB, 0, stream>>>(
      src0, dst0, ee1, s1, feat1, FEAT1, agg1, FEAT1, E0, NH, HID);

  bias_elu_kernel<<<blocks_for((size_t)N1 * FEAT1), TPB, 0, stream>>>(agg1, b1, N1, FEAT1);

  // ---------------- Layer 2 ----------------
  pad_w2_kernel<<<blocks_for((size_t)F_IN * FEAT2_PAD), TPB, 0, stream>>>(W2, Bp);
  {
    int tiles = (N1 / 16) * (FEAT2_PAD / 64);
    wmma_gemm_f32_kernel<<<(tiles + 7) / 8, TPB, 0, stream>>>(agg1, Bp, feat2, N1, FEAT2_PAD, F_IN);
  }
  attn_coeff_kernel<<<blocks_for((size_t)N1 * NH), TPB, 0, stream>>>(
      feat2, FEAT2_PAD, al2, ar2, el2, er2, N1, N2, NH, NC);

  fill_u32_kernel<<<blocks_for((size_t)N2 * NH), TPB, 0, stream>>>(m2, NEG_INF_KEY, (size_t)N2 * NH);
  fill_f32_kernel<<<blocks_for((size_t)N2 * NH), TPB, 0, stream>>>(s2, 0.f, (size_t)N2 * NH);
  fill_f32_kernel<<<blocks_for((size_t)N2 * FEAT2), TPB, 0, stream>>>(agg2, 0.f, (size_t)N2 * FEAT2);

  edge_logit_max_kernel<<<blocks_for((size_t)E1 * NH), TPB, 0, stream>>>(
      src1, dst1, el2, er2, ee2, m2, E1, NH);
  edge_exp_sum_kernel<<<blocks_for((size_t)E1 * NH), TPB, 0, stream>>>(
      dst1, ee2, m2, s2, E1, NH);
  edge_aggregate_kernel<<<blocks_for((size_t)E1 * FEAT2), TPB, 0, stream>>>(
      src1, dst1, ee2, s2, feat2, FEAT2_PAD, agg2, FEAT2, E1, NH, NC);

  head_mean_kernel<<<blocks_for((size_t)N2 * NC), TPB, 0, stream>>>(agg2, b2, out);
}